// RadialAngularEmbedding_52707838657163
// MI455X (gfx1250) — compile-verified
//
#include <hip/hip_runtime.h>
#include <hip/hip_bf16.h>

#define N_NODES 20000
#define N_EDGES 320000
#define NCH 64
#define NBESSEL 8
#define HID 6
#define SH_DIM 9
#define MSG_ROW 576          // 64*9 floats per node
#define LDS_ROW 580          // padded row stride (floats): conflict-free, 16B aligned

typedef float v2f __attribute__((ext_vector_type(2)));
typedef float v8f __attribute__((ext_vector_type(8)));

__device__ __forceinline__ float silu(float x) {
    return x / (1.0f + __expf(-x));
}

// ---------------- Kernel 0: zero the msg accumulator ----------------
__global__ void zero_ws_kernel(float* __restrict__ p, int n) {
    int i = blockIdx.x * blockDim.x + threadIdx.x;
    if (i < n) p[i] = 0.0f;
}

// ---------------- Kernel 1: radial MLP + tensor product + atomic scatter ----
// Block = (64, 4): threadIdx.x = channel c, threadIdx.y selects edge. Each
// wave32 spans lanes of a single edge -> edge-dependent values scalarize.
__global__ __launch_bounds__(256) void edge_scatter_kernel(
    const float* __restrict__ length,     // (E, 8)
    const float* __restrict__ node_feat,  // (N, 64)
    const float* __restrict__ sh,         // (E, 9)
    const int*   __restrict__ edge_index, // (2, E)
    const float* __restrict__ W1,         // (8, 6)
    const float* __restrict__ W2,         // (6, 6)
    const float* __restrict__ W3,         // (6, 6)
    const float* __restrict__ W4,         // (6, 192)
    float*       __restrict__ msg)        // (N, 64, 9) accumulator
{
    const int c = threadIdx.x;                       // channel 0..63
    const int e = blockIdx.x * 4 + threadIdx.y;      // edge id (grid exact)

    const float inv_sqrt_nb = 0.35355339059327373f;  // 1/sqrt(8)
    const float inv_sqrt_h  = 0.40824829046386307f;  // 1/sqrt(6)

    // ---- radial MLP (wave-uniform) ----
    float h0[HID], h1[HID], h2[HID];
    #pragma unroll
    for (int j = 0; j < HID; ++j) {
        float acc = 0.0f;
        #pragma unroll
        for (int k = 0; k < NBESSEL; ++k)
            acc += length[e * NBESSEL + k] * W1[k * HID + j];
        h0[j] = silu(acc * inv_sqrt_nb);
    }
    #pragma unroll
    for (int j = 0; j < HID; ++j) {
        float acc = 0.0f;
        #pragma unroll
        for (int k = 0; k < HID; ++k)
            acc += h0[k] * W2[k * HID + j];
        h1[j] = silu(acc * inv_sqrt_h);
    }
    #pragma unroll
    for (int j = 0; j < HID; ++j) {
        float acc = 0.0f;
        #pragma unroll
        for (int k = 0; k < HID; ++k)
            acc += h1[k] * W3[k * HID + j];
        h2[j] = silu(acc * inv_sqrt_h);
    }

    // ---- per-channel tensor-product weights: tp_w[e, l, c] ----
    float wl[3];
    #pragma unroll
    for (int l = 0; l < 3; ++l) {
        float acc = 0.0f;
        #pragma unroll
        for (int j = 0; j < HID; ++j)
            acc += h2[j] * W4[j * (3 * NCH) + l * NCH + c];
        wl[l] = acc * inv_sqrt_h;
    }

    const int snd = edge_index[e];
    const int rcv = edge_index[N_EDGES + e];
    const float x = node_feat[snd * NCH + c];

    float* base = msg + ((size_t)rcv * NCH + c) * SH_DIM;
    const int lmap[SH_DIM] = {0, 1, 1, 1, 2, 2, 2, 2, 2};
    #pragma unroll
    for (int m = 0; m < SH_DIM; ++m) {
        float v = x * sh[e * SH_DIM + m] * wl[lmap[m]];
        atomicAdd(base + m, v);   // -> global_atomic_add_f32 (L2-resident)
    }
}

// -------- async memory->LDS copy (CDNA5 GLOBAL_LOAD_ASYNC_TO_LDS_B128) ------
__device__ __forceinline__ void async_copy_b128(unsigned lds_off, const void* gptr) {
    unsigned long long ga = (unsigned long long)(uintptr_t)gptr;
    asm volatile("global_load_async_to_lds_b128 %0, %1, off"
                 :: "v"(lds_off), "v"(ga) : "memory");
}

// ---------------- Kernel 2: per-m channel mixing via f32 WMMA ----------------
// out[n, d, m] = (1/8) * sum_c Wlin[lmap[m]][d, c] * msg[n, c, m]
// One 128-thread block owns a 16-node slab: async-stage msg[16, 576] into LDS
// once (coalesced b128), then 4 waves x 9 m's x 16 K-steps of
// V_WMMA_F32_16X16X4_F32 read A from LDS (36x reuse of the staged slab).
// Grid exact (20000/16 blocks) -> EXEC all 1s for every wave.
__global__ __launch_bounds__(128) void node_linear_wmma_kernel(
    const float* __restrict__ msg,   // (N, 64, 9)
    const float* __restrict__ Wlin,  // (3, 64, 64)
    float*       __restrict__ out)   // (N, 64, 9)
{
    __shared__ float smem[16 * LDS_ROW];   // 37,120 B of the 320 KB WGP LDS

    const int tid   = threadIdx.x;
    const int nbase = blockIdx.x * 16;

    // ---- stage 16 x 576 floats: 2304 float4's, 18 rounds x 128 threads ----
    const float* gbase = msg + (size_t)nbase * MSG_ROW;
    #pragma unroll
    for (int it = 0; it < 18; ++it) {
        const int idx = it * 128 + tid;        // float4 index 0..2303
        const int row = idx / 144;             // 144 float4 per node row
        const int col = idx - row * 144;
        const unsigned lds_off =
            (unsigned)(uintptr_t)&smem[row * LDS_ROW + col * 4];
        async_copy_b128(lds_off, gbase + row * MSG_ROW + col * 4);
    }
    asm volatile("s_wait_asynccnt 0x0" ::: "memory");
    __syncthreads();

    const int lane  = tid & 31;
    const int wave  = tid >> 5;            // d-tile 0..3
    const int dbase = wave * 16;
    const int row   = lane & 15;           // A: M index; B: N index
    const int khalf = (lane >> 4) * 2;     // lanes 16-31 hold K+2/K+3
    const int mext  = (lane >> 4) * 8;

    for (int m = 0; m < SH_DIM; ++m) {
        const int l = (m == 0) ? 0 : ((m < 4) ? 1 : 2);   // lmap[m]
        const float* Wl = Wlin + l * NCH * NCH;

        v8f acc = {};
        #pragma unroll
        for (int kk = 0; kk < 16; ++kk) {
            const int k0 = kk * 4 + khalf;
            v2f a, b;
            // A(16x4): A[M=row][K] = msg[nbase+row, K, m]  (from LDS)
            a.x = smem[row * LDS_ROW + (k0 + 0) * SH_DIM + m];
            a.y = smem[row * LDS_ROW + (k0 + 1) * SH_DIM + m];
            // B(4x16): B[K][N=row] = Wlin[l][dbase+row, K]
            const size_t brow = (size_t)(dbase + row) * NCH;
            b.x = Wl[brow + k0 + 0];
            b.y = Wl[brow + k0 + 1];
            acc = __builtin_amdgcn_wmma_f32_16x16x4_f32(
                /*neg_a=*/false, a, /*neg_b=*/false, b,
                /*c_mod=*/(short)0, acc, /*reuse_a=*/false, /*reuse_b=*/false);
        }

        // C/D: VGPR r -> lanes 0-15: M=r, N=lane; lanes 16-31: M=r+8, N=lane-16
        #pragma unroll
        for (int r = 0; r < 8; ++r) {
            const int node = nbase + r + mext;
            out[(size_t)node * MSG_ROW + (size_t)(dbase + row) * SH_DIM + m] =
                acc[r] * 0.125f;
        }
    }
}

// ---------------------------------------------------------------------------
extern "C" void kernel_launch(void* const* d_in, const int* in_sizes, int n_in,
                              void* d_out, int out_size, void* d_ws, size_t ws_size,
                              hipStream_t stream) {
    const float* length          = (const float*)d_in[0];
    const float* node_features   = (const float*)d_in[1];
    const float* edge_attributes = (const float*)d_in[2];
    const int*   edge_index      = (const int*)d_in[3];
    const float* W1              = (const float*)d_in[4];
    const float* W2              = (const float*)d_in[5];
    const float* W3              = (const float*)d_in[6];
    const float* W4              = (const float*)d_in[7];
    const float* Wlin            = (const float*)d_in[8];
    float*       out             = (float*)d_out;
    float*       msg             = (float*)d_ws;   // (N, 64, 9) accumulator

    const int msg_elems = N_NODES * NCH * SH_DIM;  // 11.52M floats = 46.08 MB

    zero_ws_kernel<<<(msg_elems + 255) / 256, 256, 0, stream>>>(msg, msg_elems);

    dim3 eb(64, 4);
    edge_scatter_kernel<<<N_EDGES / 4, eb, 0, stream>>>(
        length, node_features, edge_attributes, edge_index,
        W1, W2, W3, W4, msg);

    node_linear_wmma_kernel<<<N_NODES / 16, 128, 0, stream>>>(msg, Wlin, out);
}